// MambaBlock_17214228922832
// MI455X (gfx1250) — compile-verified
//
#include <hip/hip_runtime.h>
#include <math.h>

#define D_MODEL 1024
#define D_STATE 16
#define D_CONV  4
#define D_INNER 2048
#define DT_RANK 64
#define NB 2
#define NL 2048
#define NTOK (NB * NL)            // 4096 tokens
#define XZ_W (2 * D_INNER)        // 4096
#define XDB_W (DT_RANK + 2 * D_STATE) // 96
#define LN_EPS 1e-5f

typedef __attribute__((ext_vector_type(2))) float v2f;
typedef __attribute__((ext_vector_type(4))) float v4f;
typedef __attribute__((ext_vector_type(8))) float v8f;

// ---------------------------------------------------------------------------
// LayerNorm: one block per token, wave32 shuffle + LDS reduction
// ---------------------------------------------------------------------------
__global__ void layernorm_kernel(const float* __restrict__ x,
                                 const float* __restrict__ w,
                                 const float* __restrict__ b,
                                 float* __restrict__ out) {
  __shared__ float s1[32], s2[32];
  const int token = blockIdx.x;
  const float* xr = x + (size_t)token * D_MODEL;
  float* orow = out + (size_t)token * D_MODEL;

  float s = 0.f, ss = 0.f;
  for (int i = threadIdx.x; i < D_MODEL; i += blockDim.x) {
    float v = xr[i];
    s += v;
    ss += v * v;
  }
  for (int off = 16; off; off >>= 1) {
    s  += __shfl_xor(s, off, 32);
    ss += __shfl_xor(ss, off, 32);
  }
  const int wid = threadIdx.x >> 5, lane = threadIdx.x & 31;
  if (lane == 0) { s1[wid] = s; s2[wid] = ss; }
  __syncthreads();
  const int nw = blockDim.x >> 5;
  if (wid == 0) {
    s  = (lane < nw) ? s1[lane] : 0.f;
    ss = (lane < nw) ? s2[lane] : 0.f;
    for (int off = 16; off; off >>= 1) {
      s  += __shfl_xor(s, off, 32);
      ss += __shfl_xor(ss, off, 32);
    }
    if (lane == 0) { s1[0] = s; s2[0] = ss; }
  }
  __syncthreads();
  const float mu  = s1[0] * (1.f / D_MODEL);
  const float var = s2[0] * (1.f / D_MODEL) - mu * mu;
  const float inv = rsqrtf(var + LN_EPS);
  for (int i = threadIdx.x; i < D_MODEL; i += blockDim.x)
    orow[i] = (xr[i] - mu) * inv * w[i] + b[i];
}

// ---------------------------------------------------------------------------
// fp32 WMMA GEMM: C[M,N] = A[M,K] * W[N,K]^T  (+ epilogue)
//   EPI 0: plain store
//   EPI 1: v = softplus(v + bias[col])      (dt_proj -> delta)
//   EPI 2: v = v + resid[row*ldc + col]     (out_proj residual add)
// One wave computes a (16*MT) x (16*NT) tile with V_WMMA_F32_16X16X4_F32.
// Software-pipelined: fragments for k+4 are loaded while wmmas consume k.
// 32-bit element offsets from uniform bases -> saddr+voffset loads.
// ---------------------------------------------------------------------------
template <int MT, int NT, int EPI>
__global__ void wmma_gemm_f32(const float* __restrict__ A, int lda,
                              const float* __restrict__ W, int ldw,
                              float* __restrict__ C, int ldc,
                              int M, int N, int K,
                              const float* __restrict__ bias,
                              const float* __restrict__ resid) {
  const int lane  = threadIdx.x & 31;
  const int wave  = threadIdx.x >> 5;
  const int gwave = blockIdx.x * (blockDim.x >> 5) + wave;
  const int nSuper = N / (16 * NT);
  const int totalWaves = (M / (16 * MT)) * nSuper;
  if (gwave >= totalWaves) return;   // wave-uniform: EXEC stays all-ones

  const int row0 = (gwave / nSuper) * (16 * MT);
  const int col0 = (gwave % nSuper) * (16 * NT);
  const int half = lane >> 4;        // selects K pair {0,1} vs {2,3}
  const int lm   = lane & 15;        // M index for A, N index for B

  const v8f zero = {0.f, 0.f, 0.f, 0.f, 0.f, 0.f, 0.f, 0.f};
  v8f acc[MT][NT];
#pragma unroll
  for (int mi = 0; mi < MT; ++mi)
#pragma unroll
    for (int j = 0; j < NT; ++j) acc[mi][j] = zero;

  unsigned aOff[MT], bOff[NT];
#pragma unroll
  for (int mi = 0; mi < MT; ++mi)
    aOff[mi] = (unsigned)(row0 + 16 * mi + lm) * (unsigned)lda + 2u * half;
#pragma unroll
  for (int j = 0; j < NT; ++j)
    bOff[j] = (unsigned)(col0 + 16 * j + lm) * (unsigned)ldw + 2u * half;

  // prologue fragments (k = 0)
  v2f a[MT], b[NT];
#pragma unroll
  for (int mi = 0; mi < MT; ++mi) a[mi] = *(const v2f*)(A + aOff[mi]);
#pragma unroll
  for (int j = 0; j < NT; ++j)    b[j]  = *(const v2f*)(W + bOff[j]);

  int k = 0;
  for (; k + 4 < K; k += 4) {
    v2f an[MT], bn[NT];
#pragma unroll
    for (int mi = 0; mi < MT; ++mi)
      an[mi] = *(const v2f*)(A + aOff[mi] + (unsigned)(k + 4));
#pragma unroll
    for (int j = 0; j < NT; ++j)
      bn[j]  = *(const v2f*)(W + bOff[j] + (unsigned)(k + 4));
#pragma unroll
    for (int mi = 0; mi < MT; ++mi)
#pragma unroll
      for (int j = 0; j < NT; ++j)
        acc[mi][j] = __builtin_amdgcn_wmma_f32_16x16x4_f32(
            false, a[mi], false, b[j], (short)0, acc[mi][j], false, false);
#pragma unroll
    for (int mi = 0; mi < MT; ++mi) a[mi] = an[mi];
#pragma unroll
    for (int j = 0; j < NT; ++j)    b[j]  = bn[j];
  }
  // epilogue K-step
#pragma unroll
  for (int mi = 0; mi < MT; ++mi)
#pragma unroll
    for (int j = 0; j < NT; ++j)
      acc[mi][j] = __builtin_amdgcn_wmma_f32_16x16x4_f32(
          false, a[mi], false, b[j], (short)0, acc[mi][j], false, false);

  // store + epilogue
#pragma unroll
  for (int mi = 0; mi < MT; ++mi) {
#pragma unroll
    for (int j = 0; j < NT; ++j) {
#pragma unroll
      for (int r = 0; r < 8; ++r) {
        const int rr = row0 + 16 * mi + r + 8 * half;
        const int cc = col0 + 16 * j + lm;
        float v = acc[mi][j][r];
        if (EPI == 1) {
          v += bias[cc];
          v = (v > 20.f) ? v : log1pf(expf(v));   // softplus
        }
        if (EPI == 2) {
          v += resid[(size_t)rr * ldc + cc];
        }
        C[(size_t)rr * ldc + cc] = v;
      }
    }
  }
}

// ---------------------------------------------------------------------------
// Depthwise causal conv (D_CONV=4) + bias + SiLU.
// x_in = first D_INNER columns of xz; layout (b, t, d), d contiguous.
// ---------------------------------------------------------------------------
__global__ void conv_silu_kernel(const float* __restrict__ xz,
                                 const float* __restrict__ cw,
                                 const float* __restrict__ cb,
                                 float* __restrict__ out) {
  const size_t total = (size_t)NTOK * D_INNER;
  const size_t idx = (size_t)blockIdx.x * blockDim.x + threadIdx.x;
  if (idx >= total) return;
  const int d = (int)(idx % D_INNER);
  const size_t bt = idx / D_INNER;
  const int t = (int)(bt % NL);
  const int b = (int)(bt / NL);

  const float* base = xz + (size_t)b * NL * XZ_W + d;  // x_in[b, :, d]
  float acc = cb[d];
#pragma unroll
  for (int j = 0; j < D_CONV; ++j) {
    const int tt = t - (D_CONV - 1) + j;
    if (tt >= 0) acc += cw[d * D_CONV + j] * base[(size_t)tt * XZ_W];
  }
  out[idx] = acc / (1.f + expf(-acc));  // SiLU
}

// ---------------------------------------------------------------------------
// Selective scan: one thread per (b, d) channel, 16 states in registers.
// u (= conv output) is overwritten in place with the gated y.
// B/C rows fetched as 8 x b128; t+16 rows prefetched (global_prefetch_b8).
// ---------------------------------------------------------------------------
#define SCAN_PF 16
__global__ void scan_kernel(const float* __restrict__ delta,
                            float* __restrict__ u_y,        // in: u, out: y
                            const float* __restrict__ xdb,  // (NTOK, 96)
                            const float* __restrict__ xz,   // z = cols [2048,4096)
                            const float* __restrict__ A_log,
                            const float* __restrict__ D_param) {
  const int gid = blockIdx.x * blockDim.x + threadIdx.x;  // 0 .. NB*D_INNER
  const int d = gid % D_INNER;
  const int b = gid / D_INNER;

  float Ad[D_STATE];
#pragma unroll
  for (int n = 0; n < D_STATE; ++n) Ad[n] = -expf(A_log[d * D_STATE + n]);
  const float Dp = D_param[d];

  float h[D_STATE];
#pragma unroll
  for (int n = 0; n < D_STATE; ++n) h[n] = 0.f;

  const float* dlt = delta + (size_t)b * NL * D_INNER + d;
  float* uy = u_y + (size_t)b * NL * D_INNER + d;
  const float* zc = xz + (size_t)b * NL * XZ_W + D_INNER + d;
  const float* bc = xdb + (size_t)b * NL * XDB_W + DT_RANK; // Bs,Cs base (16B aligned)

  for (int t = 0; t < NL; ++t) {
    // prefetch ahead to hide serial-loop latency
    if (t + SCAN_PF < NL) {
      __builtin_prefetch(dlt + (size_t)(t + SCAN_PF) * D_INNER, 0, 1);
      __builtin_prefetch(uy  + (size_t)(t + SCAN_PF) * D_INNER, 0, 1);
      __builtin_prefetch(zc  + (size_t)(t + SCAN_PF) * XZ_W,    0, 1);
      __builtin_prefetch(bc  + (size_t)(t + SCAN_PF) * XDB_W,   0, 1);
    }
    const float dt = dlt[(size_t)t * D_INNER];
    const float u  = uy[(size_t)t * D_INNER];
    const float du = dt * u;
    const v4f* bcr4 = (const v4f*)(bc + (size_t)t * XDB_W);
    v4f Bv[4], Cv[4];
#pragma unroll
    for (int q = 0; q < 4; ++q) Bv[q] = bcr4[q];
#pragma unroll
    for (int q = 0; q < 4; ++q) Cv[q] = bcr4[4 + q];
    float y = 0.f;
#pragma unroll
    for (int n = 0; n < D_STATE; ++n) {
      const float dA = expf(dt * Ad[n]);
      h[n] = dA * h[n] + du * Bv[n >> 2][n & 3];   // Bs[b,t,n]
      y += h[n] * Cv[n >> 2][n & 3];               // Cs[b,t,n]
    }
    y += u * Dp;
    const float z = zc[(size_t)t * XZ_W];
    y *= z / (1.f + expf(-z));                     // * silu(z)
    uy[(size_t)t * D_INNER] = y;
  }
}

// ---------------------------------------------------------------------------
extern "C" void kernel_launch(void* const* d_in, const int* in_sizes, int n_in,
                              void* d_out, int out_size, void* d_ws, size_t ws_size,
                              hipStream_t stream) {
  const float* x         = (const float*)d_in[0];
  const float* norm_w    = (const float*)d_in[1];
  const float* norm_b    = (const float*)d_in[2];
  const float* in_proj_w = (const float*)d_in[3];  // (4096, 1024)
  const float* conv_w    = (const float*)d_in[4];  // (2048, 1, 4)
  const float* conv_b    = (const float*)d_in[5];
  const float* x_proj_w  = (const float*)d_in[6];  // (96, 2048)
  const float* dt_proj_w = (const float*)d_in[7];  // (2048, 64)
  const float* dt_proj_b = (const float*)d_in[8];
  const float* A_log     = (const float*)d_in[9];  // (2048, 16)
  const float* D_param   = (const float*)d_in[10];
  const float* out_proj_w= (const float*)d_in[11]; // (1024, 2048)
  float* out = (float*)d_out;

  float* ws    = (float*)d_ws;
  float* xn    = ws;                                  // 4096*1024
  float* xz    = xn    + (size_t)NTOK * D_MODEL;      // 4096*4096
  float* xconv = xz    + (size_t)NTOK * XZ_W;         // 4096*2048 (becomes y)
  float* xdb   = xconv + (size_t)NTOK * D_INNER;      // 4096*96
  float* delta = xdb   + (size_t)NTOK * XDB_W;        // 4096*2048

  // 1) LayerNorm
  layernorm_kernel<<<NTOK, 256, 0, stream>>>(x, norm_w, norm_b, xn);

  // 2) in_proj: xz = xn @ in_proj_w^T   (M=4096, N=4096, K=1024)
  {
    int waves = (NTOK / 32) * (XZ_W / 64);
    wmma_gemm_f32<2, 4, 0><<<(waves + 7) / 8, 256, 0, stream>>>(
        xn, D_MODEL, in_proj_w, D_MODEL, xz, XZ_W, NTOK, XZ_W, D_MODEL,
        nullptr, nullptr);
  }

  // 3) depthwise causal conv + SiLU
  {
    size_t total = (size_t)NTOK * D_INNER;
    conv_silu_kernel<<<(unsigned)((total + 255) / 256), 256, 0, stream>>>(
        xz, conv_w, conv_b, xconv);
  }

  // 4) x_proj: xdb = xconv @ x_proj_w^T (M=4096, N=96, K=2048)
  {
    int waves = (NTOK / 32) * (XDB_W / 32);
    wmma_gemm_f32<2, 2, 0><<<(waves + 7) / 8, 256, 0, stream>>>(
        xconv, D_INNER, x_proj_w, D_INNER, xdb, XDB_W, NTOK, XDB_W, D_INNER,
        nullptr, nullptr);
  }

  // 5) dt_proj + softplus: delta = softplus(xdb[:, :64] @ dt_proj_w^T + b)
  {
    int waves = (NTOK / 32) * (D_INNER / 64);
    wmma_gemm_f32<2, 4, 1><<<(waves + 7) / 8, 256, 0, stream>>>(
        xdb, XDB_W, dt_proj_w, DT_RANK, delta, D_INNER, NTOK, D_INNER, DT_RANK,
        dt_proj_b, nullptr);
  }

  // 6) selective scan, fused u*D + silu(z) gating; y overwrites xconv
  scan_kernel<<<(NB * D_INNER) / 256, 256, 0, stream>>>(
      delta, xconv, xdb, xz, A_log, D_param);

  // 7) out_proj + residual: out = x + y @ out_proj_w^T (M=4096, N=1024, K=2048)
  {
    int waves = (NTOK / 32) * (D_MODEL / 64);
    wmma_gemm_f32<2, 4, 2><<<(waves + 7) / 8, 256, 0, stream>>>(
        xconv, D_INNER, out_proj_w, D_INNER, out, D_MODEL, NTOK, D_MODEL,
        D_INNER, nullptr, x);
  }
}